// MuleDetectorGNN_81011673137241
// MI455X (gfx1250) — compile-verified
//
#include <hip/hip_runtime.h>
#include <hip/hip_bf16.h>

#define NN   50000
#define EE   1600000
#define DIN  32
#define DH   64
#define HH   4
#define CC   16
#define DEE  5
#define LLAYERS 3
#define ETOT (EE + NN)
#define NEG_SLOPE 0.2f

typedef __attribute__((ext_vector_type(2))) float v2f;
typedef __attribute__((ext_vector_type(8))) float v8f;

// ---- ordered-float encoding for atomic max over signed floats ----
__device__ __forceinline__ unsigned f2ord(float f) {
    unsigned u = __float_as_uint(f);
    return (u & 0x80000000u) ? ~u : (u | 0x80000000u);
}
__device__ __forceinline__ float ord2f(unsigned u) {
    return (u & 0x80000000u) ? __uint_as_float(u & 0x7FFFFFFFu)
                             : __uint_as_float(~u);
}

// ---- self-loop edge_attr mean: scatter sums + counts ----
__global__ void k_scatter_ea(const int* __restrict__ ei, const float* __restrict__ ea,
                             float* __restrict__ sum, float* __restrict__ cnt) {
    int e = blockIdx.x * blockDim.x + threadIdx.x;
    if (e >= EE) return;
    int dst = ei[EE + e];
#pragma unroll
    for (int j = 0; j < DEE; ++j)
        atomicAdd(&sum[(size_t)dst * DEE + j], ea[(size_t)e * DEE + j]);
    atomicAdd(&cnt[dst], 1.0f);
}

__global__ void k_mean_finalize(float* __restrict__ sum, const float* __restrict__ cnt) {
    int i = blockIdx.x * blockDim.x + threadIdx.x;
    if (i >= NN * DEE) return;
    int n = i / DEE;
    sum[i] = sum[i] / fmaxf(cnt[n], 1.0f);
}

// ---- f32 WMMA GEMM: C[NN x DH] = A[NN x K] * B[K x DH] (+ bias) ----
// one wave per 16x16 output tile; K multiple of 4; NN multiple of 16.
__global__ void k_gemm_wmma(const float* __restrict__ A, const float* __restrict__ B,
                            const float* __restrict__ bias, float* __restrict__ C,
                            int K) {
    int wid  = (blockIdx.x * blockDim.x + threadIdx.x) >> 5;
    const int tiles_col = DH / 16;                    // 4
    int tile_m = wid / tiles_col;
    int tile_n = wid - tile_m * tiles_col;
    if (tile_m * 16 >= NN) return;                    // uniform per wave
    int lane = threadIdx.x & 31;
    int half = lane >> 4;                             // 0: K pair {0,1}; 1: {2,3}
    int l16  = lane & 15;
    const float* Arow = A + (size_t)(tile_m * 16 + l16) * K;
    int colbase = tile_n * 16 + l16;
    v8f acc = {};
    for (int k = 0; k < K; k += 4) {
        int ka = k + half * 2;
        v2f av, bv;
        av[0] = Arow[ka];
        av[1] = Arow[ka + 1];
        bv[0] = B[(size_t)ka * DH + colbase];
        bv[1] = B[(size_t)(ka + 1) * DH + colbase];
        acc = __builtin_amdgcn_wmma_f32_16x16x4_f32(
            /*neg_a=*/false, av, /*neg_b=*/false, bv,
            /*c_mod=*/(short)0, acc, /*reuse_a=*/false, /*reuse_b=*/false);
    }
    float bb = bias ? bias[colbase] : 0.0f;
#pragma unroll
    for (int v = 0; v < 8; ++v) {
        int row = tile_m * 16 + v + half * 8;
        C[(size_t)row * DH + colbase] = acc[v] + bb;
    }
}

// ---- per-node attention logits a_s, a_d from projected features ----
__global__ void k_as_ad(const float* __restrict__ hp, const float* __restrict__ att_s,
                        const float* __restrict__ att_d,
                        float* __restrict__ a_s, float* __restrict__ a_d) {
    int n = blockIdx.x * blockDim.x + threadIdx.x;
    if (n >= NN) return;
    const float* hr = hp + (size_t)n * DH;
#pragma unroll
    for (int h = 0; h < HH; ++h) {
        float ss = 0.f, sd = 0.f;
#pragma unroll
        for (int c = 0; c < CC; ++c) {
            float v = hr[h * CC + c];
            ss += v * att_s[h * CC + c];
            sd += v * att_d[h * CC + c];
        }
        a_s[n * HH + h] = ss;
        a_d[n * HH + h] = sd;
    }
}

// ---- fold lin_edge @ att_edge into w_eff[DE][H] (exact: a_e is linear in ea) ----
__global__ void k_weff(const float* __restrict__ lin_e, const float* __restrict__ att_e,
                       float* __restrict__ w_eff) {
    int i = threadIdx.x;
    if (i >= DEE * HH) return;
    int de = i / HH, h = i - de * HH;
    float s = 0.f;
#pragma unroll
    for (int c = 0; c < CC; ++c) s += lin_e[de * DH + h * CC + c] * att_e[h * CC + c];
    w_eff[i] = s;
}

// ---- edge logits + leaky relu + segment max (ordered-uint atomicMax) ----
__global__ void k_alpha(const int* __restrict__ ei, const float* __restrict__ ea,
                        const float* __restrict__ mean_ea,
                        const float* __restrict__ a_s, const float* __restrict__ a_d,
                        const float* __restrict__ w_eff,
                        float* __restrict__ alpha, unsigned* __restrict__ m_u) {
    int e = blockIdx.x * blockDim.x + threadIdx.x;
    if (e >= ETOT) return;
    int src, dst;
    const float* eap;
    if (e < EE) { src = ei[e]; dst = ei[EE + e]; eap = ea + (size_t)e * DEE; }
    else        { src = dst = e - EE;            eap = mean_ea + (size_t)(e - EE) * DEE; }
    float ev[DEE];
#pragma unroll
    for (int j = 0; j < DEE; ++j) ev[j] = eap[j];
#pragma unroll
    for (int h = 0; h < HH; ++h) {
        float ae = 0.f;
#pragma unroll
        for (int j = 0; j < DEE; ++j) ae += ev[j] * w_eff[j * HH + h];
        float a = a_s[src * HH + h] + a_d[dst * HH + h] + ae;
        a = (a > 0.f) ? a : NEG_SLOPE * a;
        alpha[(size_t)e * HH + h] = a;
        atomicMax(&m_u[dst * HH + h], f2ord(a));
    }
}

// ---- exp(alpha - max) + segment sum ----
__global__ void k_exp(const int* __restrict__ ei, const unsigned* __restrict__ m_u,
                      float* __restrict__ alpha, float* __restrict__ denom) {
    int idx = blockIdx.x * blockDim.x + threadIdx.x;
    if (idx >= ETOT * HH) return;
    int e = idx >> 2, h = idx & 3;
    int dst = (e < EE) ? ei[EE + e] : (e - EE);
    float m  = ord2f(m_u[dst * HH + h]);
    float ex = expf(alpha[idx] - m);
    alpha[idx] = ex;
    atomicAdd(&denom[dst * HH + h], ex);
}

// ---- weighted aggregation: one wave per edge, lane -> channels {l, l+32} ----
__global__ void k_aggr(const int* __restrict__ ei, const float* __restrict__ hp,
                       const float* __restrict__ ex, const float* __restrict__ denom,
                       float* __restrict__ hout) {
    int t = blockIdx.x * blockDim.x + threadIdx.x;
    int e = t >> 5;
    if (e >= ETOT) return;
    int lane = t & 31;
    int src, dst;
    if (e < EE) { src = ei[e]; dst = ei[EE + e]; }
    else        { src = dst = e - EE; }
    int c0 = lane, c1 = lane + 32;
    int h0 = c0 >> 4, h1 = c1 >> 4;
    float w0 = ex[(size_t)e * HH + h0] / denom[dst * HH + h0];
    float w1 = ex[(size_t)e * HH + h1] / denom[dst * HH + h1];
    const float* hr = hp + (size_t)src * DH;
    atomicAdd(&hout[(size_t)dst * DH + c0], hr[c0] * w0);
    atomicAdd(&hout[(size_t)dst * DH + c1], hr[c1] * w1);
}

// ---- + bias, optional ELU ----
__global__ void k_final(float* __restrict__ h, const float* __restrict__ bias, int do_elu) {
    int i = blockIdx.x * blockDim.x + threadIdx.x;
    if (i >= NN * DH) return;
    float v = h[i] + bias[i & (DH - 1)];
    if (do_elu) v = (v > 0.f) ? v : (expf(v) - 1.0f);
    h[i] = v;
}

// ---- classifier + risk heads + embedding copy ----
__global__ void k_heads(const float* __restrict__ h,
                        const float* __restrict__ w1, const float* __restrict__ b1,
                        const float* __restrict__ w2, const float* __restrict__ b2,
                        const float* __restrict__ w3, const float* __restrict__ b3,
                        const float* __restrict__ rw1, const float* __restrict__ rb1,
                        const float* __restrict__ rw2, const float* __restrict__ rb2,
                        float* __restrict__ out) {
    int n = blockIdx.x * blockDim.x + threadIdx.x;
    if (n >= NN) return;
    const float* hr = h + (size_t)n * DH;
    float hv[DH];
#pragma unroll
    for (int c = 0; c < DH; ++c) hv[c] = hr[c];

    float c1[DH / 2];
    for (int j = 0; j < DH / 2; ++j) {
        float s = b1[j];
        for (int c = 0; c < DH; ++c) s += hv[c] * w1[c * (DH / 2) + j];
        c1[j] = fmaxf(s, 0.f);
    }
    float c2[DH / 4];
    for (int j = 0; j < DH / 4; ++j) {
        float s = b2[j];
        for (int c = 0; c < DH / 2; ++c) s += c1[c] * w2[c * (DH / 4) + j];
        c2[j] = fmaxf(s, 0.f);
    }
    float p = b3[0];
    for (int c = 0; c < DH / 4; ++c) p += c2[c] * w3[c];
    p = 1.0f / (1.0f + expf(-p));

    float r1[DH / 2];
    for (int j = 0; j < DH / 2; ++j) {
        float s = rb1[j];
        for (int c = 0; c < DH; ++c) s += hv[c] * rw1[c * (DH / 2) + j];
        r1[j] = fmaxf(s, 0.f);
    }
    float r = rb2[0];
    for (int c = 0; c < DH / 2; ++c) r += r1[c] * rw2[c];
    r = 100.0f / (1.0f + expf(-r));

    out[n] = p;
    out[NN + n] = r;
#pragma unroll
    for (int c = 0; c < DH; ++c) out[2 * (size_t)NN + (size_t)n * DH + c] = hv[c];
}

static inline int cdiv(long long a, int b) { return (int)((a + b - 1) / b); }

extern "C" void kernel_launch(void* const* d_in, const int* in_sizes, int n_in,
                              void* d_out, int out_size, void* d_ws, size_t ws_size,
                              hipStream_t stream) {
    const float* x        = (const float*)d_in[0];
    const int*   ei       = (const int*)  d_in[1];
    const float* eattr    = (const float*)d_in[2];
    const float* in_w     = (const float*)d_in[3];
    const float* in_b     = (const float*)d_in[4];
    const float* gat_lin  = (const float*)d_in[5];
    const float* att_src  = (const float*)d_in[6];
    const float* att_dst  = (const float*)d_in[7];
    const float* lin_edge = (const float*)d_in[8];
    const float* att_edge = (const float*)d_in[9];
    const float* gat_bias = (const float*)d_in[10];
    const float* cls_w1 = (const float*)d_in[11]; const float* cls_b1 = (const float*)d_in[12];
    const float* cls_w2 = (const float*)d_in[13]; const float* cls_b2 = (const float*)d_in[14];
    const float* cls_w3 = (const float*)d_in[15]; const float* cls_b3 = (const float*)d_in[16];
    const float* rk_w1  = (const float*)d_in[17]; const float* rk_b1  = (const float*)d_in[18];
    const float* rk_w2  = (const float*)d_in[19]; const float* rk_b2  = (const float*)d_in[20];
    float* out = (float*)d_out;

    // workspace carve-out (all 4-byte types)
    float* fws = (float*)d_ws;
    size_t o = 0;
    float* h_a     = fws + o; o += (size_t)NN * DH;
    float* h_b     = fws + o; o += (size_t)NN * DH;
    float* hp      = fws + o; o += (size_t)NN * DH;
    float* a_s     = fws + o; o += (size_t)NN * HH;
    float* a_d     = fws + o; o += (size_t)NN * HH;
    float* alpha   = fws + o; o += (size_t)ETOT * HH;
    unsigned* m_u  = (unsigned*)(fws + o); o += (size_t)NN * HH;
    float* denom   = fws + o; o += (size_t)NN * HH;
    float* mean_ea = fws + o; o += (size_t)NN * DEE;
    float* cnt     = fws + o; o += (size_t)NN;
    float* w_eff   = fws + o; o += DEE * HH;
    (void)ws_size; (void)in_sizes; (void)n_in; (void)out_size;

    const int B = 256;

    // self-loop edge_attr mean (PyG fill_value='mean')
    hipMemsetAsync(mean_ea, 0, (size_t)NN * DEE * sizeof(float), stream);
    hipMemsetAsync(cnt,     0, (size_t)NN * sizeof(float),        stream);
    k_scatter_ea<<<cdiv(EE, B), B, 0, stream>>>(ei, eattr, mean_ea, cnt);
    k_mean_finalize<<<cdiv((long long)NN * DEE, B), B, 0, stream>>>(mean_ea, cnt);

    // input projection: h_a = x @ in_w + in_b   (WMMA f32 16x16x4)
    const int gemm_waves  = (NN / 16) * (DH / 16);
    const int gemm_blocks = cdiv((long long)gemm_waves * 32, B);
    k_gemm_wmma<<<gemm_blocks, B, 0, stream>>>(x, in_w, in_b, h_a, DIN);

    float* h_cur = h_a;
    float* h_nxt = h_b;
    for (int l = 0; l < LLAYERS; ++l) {
        const float* lin  = gat_lin  + (size_t)l * DH * DH;
        const float* ats  = att_src  + (size_t)l * HH * CC;
        const float* atd  = att_dst  + (size_t)l * HH * CC;
        const float* line = lin_edge + (size_t)l * DEE * DH;
        const float* ate  = att_edge + (size_t)l * HH * CC;
        const float* gb   = gat_bias + (size_t)l * DH;

        // hp = h_cur @ lin (WMMA f32)
        k_gemm_wmma<<<gemm_blocks, B, 0, stream>>>(h_cur, lin, nullptr, hp, DH);
        k_as_ad<<<cdiv(NN, B), B, 0, stream>>>(hp, ats, atd, a_s, a_d);
        k_weff<<<1, 32, 0, stream>>>(line, ate, w_eff);

        hipMemsetAsync(m_u,   0, (size_t)NN * HH * sizeof(unsigned), stream); // 0 == -inf sentinel
        hipMemsetAsync(denom, 0, (size_t)NN * HH * sizeof(float),    stream);
        hipMemsetAsync(h_nxt, 0, (size_t)NN * DH * sizeof(float),    stream);

        k_alpha<<<cdiv(ETOT, B), B, 0, stream>>>(ei, eattr, mean_ea, a_s, a_d, w_eff, alpha, m_u);
        k_exp<<<cdiv((long long)ETOT * HH, B), B, 0, stream>>>(ei, m_u, alpha, denom);
        k_aggr<<<cdiv((long long)ETOT * 32, B), B, 0, stream>>>(ei, hp, alpha, denom, h_nxt);
        k_final<<<cdiv((long long)NN * DH, B), B, 0, stream>>>(h_nxt, gb, (l < LLAYERS - 1) ? 1 : 0);

        float* t = h_cur; h_cur = h_nxt; h_nxt = t;
    }

    k_heads<<<cdiv(NN, B), B, 0, stream>>>(h_cur,
                                           cls_w1, cls_b1, cls_w2, cls_b2, cls_w3, cls_b3,
                                           rk_w1, rk_b1, rk_w2, rk_b2, out);
}